// GAT2_53197464928925
// MI455X (gfx1250) — compile-verified
//
#include <hip/hip_runtime.h>
#include <hip/hip_bf16.h>
#include <math.h>

#define NN 50000
#define EE 800000
#define DD 64
#define LLAYERS 4
#define GG 128
#define TT 10
#define NEG_SLOPE_C 0.2f
#define EPS_C 1e-16f
// flip(-inf): bits(-inf)=0xFF800000 (negative) -> ~u = 0x007FFFFF
#define FLIP_NEG_INF 0x007FFFFFu

typedef __attribute__((ext_vector_type(2))) float v2f;
typedef __attribute__((ext_vector_type(8))) float v8f;

// ---- order-preserving float <-> uint transform (for atomicMax on floats) ----
__device__ __forceinline__ unsigned flip_f(float f) {
  unsigned u = __float_as_uint(f);
  return (u & 0x80000000u) ? ~u : (u | 0x80000000u);
}
__device__ __forceinline__ float unflip_f(unsigned u) {
  return __uint_as_float((u & 0x80000000u) ? (u & 0x7FFFFFFFu) : ~u);
}

// ---- h = node_emb[x] gather ----
__global__ void embed_k(const int* __restrict__ x, const float* __restrict__ emb,
                        float* __restrict__ h, int n) {
  int t = blockIdx.x * blockDim.x + threadIdx.x;
  if (t >= n * DD) return;
  int i = t >> 6, c = t & 63;
  h[t] = emb[x[i] * DD + c];
}

// ---- per-layer scratch init: acc=0 (N*64), denom=0, m_u=flip(-inf) (N) ----
__global__ void layer_init_k(float* __restrict__ acc, float* __restrict__ denom,
                             unsigned* __restrict__ m_u, int n) {
  int t = blockIdx.x * blockDim.x + threadIdx.x;
  if (t < n * DD) acc[t] = 0.0f;
  if (t < n) { denom[t] = 0.0f; m_u[t] = FLIP_NEG_INF; }
}

__global__ void fill_f32_k(float* __restrict__ p, float v, int n) {
  int t = blockIdx.x * blockDim.x + threadIdx.x;
  if (t < n) p[t] = v;
}

// ---- ae[a] = dot(edge_emb[a] @ W_e.T, att_edge) = dot(edge_emb[a], att_edge @ W_e) ----
__global__ void ae_tab_k(const float* __restrict__ e_emb,   // 4 x 64
                         const float* __restrict__ W_e,     // 64 x 64 (row major)
                         const float* __restrict__ a_edge,  // 64
                         float* __restrict__ ae) {          // 4
  __shared__ float v[DD];
  int k = threadIdx.x;  // 64 threads
  float s = 0.0f;
  for (int j = 0; j < DD; ++j) s += W_e[j * DD + k] * a_edge[j];
  v[k] = s;
  __syncthreads();
  if (k < 4) {
    float t = 0.0f;
    for (int q = 0; q < DD; ++q) t += e_emb[k * DD + q] * v[q];
    ae[k] = t;
  }
}

// ============================================================================
// Fused WMMA GEMM:  Y[M x NT*16] = X[M x 64] @ Wm[NT*16 x 64]^T
//   - W staged in LDS once per block (broadcast to 4 waves via ds_load)
//   - each wave owns a 16-row strip, computes all NT column tiles from one
//     A-fragment load per K-step (A reuse x NT)
//   - optional fused epilogue: bias + relu,  and  asrc=g@a_src / adst=g@a_dst
//     reduced from the live accumulators (cross-lane shfl within 16-lane half)
// Requires M % 16 == 0.
// ============================================================================
template <int NT>
__global__ void __launch_bounds__(128)
gemm_fused(const float* __restrict__ X, const float* __restrict__ Wm,
           const float* __restrict__ bias,
           const float* __restrict__ a_src, const float* __restrict__ a_dst,
           float* __restrict__ Y, float* __restrict__ asrc,
           float* __restrict__ adst, int M, int do_relu) {
  const int K = DD;
  const int Nout = NT * 16;
  __shared__ float wlds[NT * 16 * DD];

  // cooperative stage of the full weight matrix into LDS (float4 per thread)
  for (int idx = threadIdx.x * 4; idx < Nout * K; idx += 128 * 4) {
    *(float4*)&wlds[idx] = *(const float4*)&Wm[idx];
  }
  __syncthreads();

  int wave = threadIdx.x >> 5;
  int lane = threadIdx.x & 31;
  int half = lane >> 4;  // selects K pair {0,1} vs {2,3} (A/B frag layout)
  int r = lane & 15;     // A row / B output-column within tile
  int m0 = blockIdx.x * 64 + wave * 16;
  if (m0 >= M) return;  // wave-uniform tail guard (M % 16 == 0)

  const float* arow = X + (size_t)(m0 + r) * K + 2 * half;

  v8f cacc[NT];
#pragma unroll
  for (int nt = 0; nt < NT; ++nt) cacc[nt] = (v8f){};

  for (int kk = 0; kk < K; kk += 4) {
    v2f a;
    a.x = arow[kk];
    a.y = arow[kk + 1];
#pragma unroll
    for (int nt = 0; nt < NT; ++nt) {
      // B = W^T: column (nt*16 + r) of B  <-  row (nt*16 + r) of W (in LDS)
      v2f b = *(const v2f*)&wlds[(nt * 16 + r) * K + kk + 2 * half];
      cacc[nt] = __builtin_amdgcn_wmma_f32_16x16x4_f32(
          false, a, false, b, (short)0, cacc[nt], false, false);
    }
  }

  // ---- store D tiles (+bias/relu) ----
#pragma unroll
  for (int nt = 0; nt < NT; ++nt) {
    int col = nt * 16 + r;
    float bv = bias ? bias[col] : 0.0f;
#pragma unroll
    for (int v = 0; v < 8; ++v) {
      int row = m0 + v + 8 * half;  // C/D layout: VGPR v -> M = v + 8*half
      float val = cacc[nt][v] + bv;
      if (do_relu) val = fmaxf(val, 0.0f);
      Y[(size_t)row * Nout + col] = val;
    }
  }

  // ---- fused attention logits from live accumulators ----
  if (asrc) {
    float as0[NT], ad0[NT];
#pragma unroll
    for (int nt = 0; nt < NT; ++nt) {
      as0[nt] = a_src[nt * 16 + r];
      ad0[nt] = a_dst[nt * 16 + r];
    }
#pragma unroll
    for (int v = 0; v < 8; ++v) {
      float p0 = 0.0f, p1 = 0.0f;
#pragma unroll
      for (int nt = 0; nt < NT; ++nt) {
        p0 += cacc[nt][v] * as0[nt];
        p1 += cacc[nt][v] * ad0[nt];
      }
      // reduce across the 16 lanes of this half (rows v+8*half live here)
      for (int off = 8; off > 0; off >>= 1) {
        p0 += __shfl_xor(p0, off, 16);
        p1 += __shfl_xor(p1, off, 16);
      }
      if (r == 0) {
        int row = m0 + v + 8 * half;
        asrc[row] = p0;
        adst[row] = p1;
      }
    }
  }
}

// ---- alpha = leaky_relu(asrc[src]+adst[dst]+ae[attr]); segment-max via atomicMax ----
__global__ void edge_alpha_k(const int* __restrict__ src, const int* __restrict__ dst,
                             const int* __restrict__ attr,
                             const float* __restrict__ asrc, const float* __restrict__ adst,
                             const float* __restrict__ ae,
                             float* __restrict__ alpha, unsigned* __restrict__ m_u, int e) {
  int t = blockIdx.x * blockDim.x + threadIdx.x;
  if (t >= e) return;
  int d = dst[t];
  float a = asrc[src[t]] + adst[d] + ae[attr[t]];
  a = (a > 0.0f) ? a : NEG_SLOPE_C * a;
  alpha[t] = a;
  atomicMax(&m_u[d], flip_f(a));
}

// ---- ex = exp(alpha - m[dst]); denom[dst] += ex ----
__global__ void edge_exp_k(const int* __restrict__ dst, const float* __restrict__ alpha,
                           const unsigned* __restrict__ m_u,
                           float* __restrict__ ex, float* __restrict__ denom, int e) {
  int t = blockIdx.x * blockDim.x + threadIdx.x;
  if (t >= e) return;
  int d = dst[t];
  float mv = unflip_f(m_u[d]);
  if (!(mv > -INFINITY && mv < INFINITY)) mv = 0.0f;  // where(isfinite, m, 0)
  float v = __expf(alpha[t] - mv);
  ex[t] = v;
  atomicAdd(&denom[d], v);
}

// ---- acc[dst] += g[src] * (ex / (denom[dst]+EPS)); one wave per edge, float2/lane ----
__global__ void edge_aggr_k(const int* __restrict__ src, const int* __restrict__ dst,
                            const float* __restrict__ ex, const float* __restrict__ denom,
                            const float* __restrict__ g, float* __restrict__ acc, int e) {
  int we = (blockIdx.x * blockDim.x + threadIdx.x) >> 5;
  int lane = threadIdx.x & 31;
  if (we >= e) return;
  int s = src[we], d = dst[we];
  float w = ex[we] / (denom[d] + EPS_C);
  const float2* gv = (const float2*)(g + (size_t)s * DD);
  float2 val = gv[lane];
  atomicAdd(&acc[(size_t)d * DD + 2 * lane], val.x * w);
  atomicAdd(&acc[(size_t)d * DD + 2 * lane + 1], val.y * w);
}

// ---- h = relu(acc + bias) ----
__global__ void bias_relu_k(const float* __restrict__ acc, const float* __restrict__ bias,
                            float* __restrict__ h, int n) {
  int t = blockIdx.x * blockDim.x + threadIdx.x;
  if (t >= n * DD) return;
  h[t] = fmaxf(acc[t] + bias[t & 63], 0.0f);
}

// ---- pooled[batch[i]] += h[i]; one wave per node ----
__global__ void pool_k(const int* __restrict__ batch, const float* __restrict__ h,
                       float* __restrict__ pooled, int n) {
  int i = (blockIdx.x * blockDim.x + threadIdx.x) >> 5;
  int lane = threadIdx.x & 31;
  if (i >= n) return;
  int b = batch[i];
  const float2* hv = (const float2*)(h + (size_t)i * DD);
  float2 val = hv[lane];
  atomicAdd(&pooled[(size_t)b * DD + 2 * lane], val.x);
  atomicAdd(&pooled[(size_t)b * DD + 2 * lane + 1], val.y);
}

// ---- out = hidden @ W2.T + b2  (128x128 @ 128x10) ----
__global__ void final_k(const float* __restrict__ hidden, const float* __restrict__ W2,
                        const float* __restrict__ b2, float* __restrict__ out) {
  int t = blockIdx.x * blockDim.x + threadIdx.x;
  if (t >= GG * TT) return;
  int gi = t / TT, o = t % TT;
  float s = b2[o];
  for (int j = 0; j < 2 * DD; ++j) s += hidden[gi * 2 * DD + j] * W2[o * 2 * DD + j];
  out[t] = s;
}

extern "C" void kernel_launch(void* const* d_in, const int* in_sizes, int n_in,
                              void* d_out, int out_size, void* d_ws, size_t ws_size,
                              hipStream_t stream) {
  (void)in_sizes; (void)n_in; (void)out_size; (void)ws_size;
  const int* x          = (const int*)d_in[0];
  const int* edge_index = (const int*)d_in[1];
  const int* src        = edge_index;
  const int* dst        = edge_index + EE;
  const int* eattr      = (const int*)d_in[2];
  const int* batch      = (const int*)d_in[3];
  const float* node_emb = (const float*)d_in[4];
  const float* edge_embs= (const float*)d_in[5];   // L x 4 x 64
  const float* Ws       = (const float*)d_in[6];   // L x 64 x 64
  const float* W_edges  = (const float*)d_in[7];   // L x 64 x 64
  const float* att_src  = (const float*)d_in[8];   // L x 64
  const float* att_dst  = (const float*)d_in[9];   // L x 64
  const float* att_edge = (const float*)d_in[10];  // L x 64
  const float* biases   = (const float*)d_in[11];  // L x 64
  const float* W1       = (const float*)d_in[12];  // 128 x 64
  const float* b1       = (const float*)d_in[13];  // 128
  const float* W2       = (const float*)d_in[14];  // 10 x 128
  const float* b2       = (const float*)d_in[15];  // 10
  float* out = (float*)d_out;

  char* ws = (char*)d_ws;
  size_t off = 0;
  auto take = [&](size_t bytes) -> void* {
    void* p = (void*)(ws + off);
    off += (bytes + 255) & ~(size_t)255;
    return p;
  };
  float*    h      = (float*)take(sizeof(float) * NN * DD);
  float*    gbuf   = (float*)take(sizeof(float) * NN * DD);
  float*    acc    = (float*)take(sizeof(float) * NN * DD);
  float*    asrc   = (float*)take(sizeof(float) * NN);
  float*    adst   = (float*)take(sizeof(float) * NN);
  unsigned* m_u    = (unsigned*)take(sizeof(unsigned) * NN);
  float*    denom  = (float*)take(sizeof(float) * NN);
  float*    alpha  = (float*)take(sizeof(float) * EE);
  float*    exb    = (float*)take(sizeof(float) * EE);
  float*    ae     = (float*)take(sizeof(float) * 4);
  float*    pooled = (float*)take(sizeof(float) * GG * DD);
  float*    hidden = (float*)take(sizeof(float) * GG * 2 * DD);

  const int B = 256;
  int blk_nd = (NN * DD + B - 1) / B;  // N*64 element grids
  int blk_nw = (NN * 32 + B - 1) / B;  // one wave per node
  int blk_e  = (EE + B - 1) / B;       // one thread per edge
  int blk_ew = (EE * 32 + B - 1) / B;  // one wave per edge

  embed_k<<<blk_nd, B, 0, stream>>>(x, node_emb, h, NN);

  for (int l = 0; l < LLAYERS; ++l) {
    layer_init_k<<<blk_nd, B, 0, stream>>>(acc, denom, m_u, NN);
    ae_tab_k<<<1, 64, 0, stream>>>(edge_embs + l * 4 * DD, W_edges + l * DD * DD,
                                   att_edge + l * DD, ae);
    // g = h @ W^T, fused asrc/adst epilogue; 4 waves/block, 64 rows/block
    int gblk = (NN + 63) / 64;
    gemm_fused<4><<<gblk, 128, 0, stream>>>(
        h, Ws + l * DD * DD, nullptr, att_src + l * DD, att_dst + l * DD,
        gbuf, asrc, adst, NN, 0);
    edge_alpha_k<<<blk_e, B, 0, stream>>>(src, dst, eattr, asrc, adst, ae,
                                          alpha, m_u, EE);
    edge_exp_k<<<blk_e, B, 0, stream>>>(dst, alpha, m_u, exb, denom, EE);
    edge_aggr_k<<<blk_ew, B, 0, stream>>>(src, dst, exb, denom, gbuf, acc, EE);
    bias_relu_k<<<blk_nd, B, 0, stream>>>(acc, biases + l * DD, h, NN);
  }

  fill_f32_k<<<(GG * DD + B - 1) / B, B, 0, stream>>>(pooled, 0.0f, GG * DD);
  pool_k<<<blk_nw, B, 0, stream>>>(batch, h, pooled, NN);

  // hidden = relu(pooled @ W1^T + b1): M=128, Nout=128 (NT=8)
  gemm_fused<8><<<(GG + 63) / 64, 128, 0, stream>>>(
      pooled, W1, b1, nullptr, nullptr, hidden, nullptr, nullptr, GG, 1);

  final_k<<<(GG * TT + B - 1) / B, B, 0, stream>>>(hidden, W2, b2, out);
}